// RNN_17489106830112
// MI455X (gfx1250) — compile-verified
//
#include <hip/hip_runtime.h>
#include <cmath>
#include <stdint.h>

// RNN-tanh forward for B=64, T=512, I=256, H=1024 on gfx1250 (MI455X).
//   Phase 1: xp = X @ W_ih^T + b_ih  (parallel WMMA-f32 GEMM) staged into d_out[B,T,H].
//   Phase 2: 512 stream-ordered steps: out[:,t,:] = tanh(out[:,t,:] + h_{t-1} @ W_hh^T + b_hh).
//   Phase 3: hidden = states[:,T-1,:].
// GEMM uses double-buffered LDS tiles filled via global_load_async_to_lds_b128 (ASYNCcnt),
// fragments read from LDS (ds_load_b64), math in exact f32 via V_WMMA_F32_16X16X4_F32.

typedef __attribute__((ext_vector_type(2))) float v2f;
typedef __attribute__((ext_vector_type(8))) float v8f;

#define RNN_B 64
#define RNN_T 512
#define RNN_I 256
#define RNN_H 1024

// s_wait_asynccnt with the handful of immediates we need (per-wave ASYNCcnt; async loads
// complete in order, so waiting for cnt<=N with N loads issued for the *next* chunk
// guarantees the *current* chunk's data is in LDS).
template <int N> __device__ __forceinline__ void wait_asynccnt() {
    static_assert(N == 0 || N == 4 || N == 6, "unsupported wait immediate");
    if constexpr (N == 0)      asm volatile("s_wait_asynccnt 0x0" ::: "memory");
    else if constexpr (N == 4) asm volatile("s_wait_asynccnt 0x4" ::: "memory");
    else                       asm volatile("s_wait_asynccnt 0x6" ::: "memory");
}

// Per-lane 16B global -> LDS async DMA (gfx1250, tracked with ASYNCcnt).
__device__ __forceinline__ void async_b128_to_lds(unsigned lds_off, const void* g) {
    asm volatile("global_load_async_to_lds_b128 %0, %1, off"
                 :: "v"(lds_off), "v"((unsigned long long)(uintptr_t)g)
                 : "memory");
}

// C[m][n] = sum_k A[m][k] * Bn[n][k]   (Bn is N x K row-major; K a compile-time constant)
// mode 0: Out[m*strideO+n] = C + bias[n]
// mode 1: Out[m*strideO+n] = tanh(Out[m*strideO+n] + C + bias[n])
// Block: (WR*TM*16) x (WC*TN*16) with WR*WC = 8 waves (256 threads). K-chunk = 32.
template <int TM, int TN, int WR, int WC, int K>
__global__ __launch_bounds__(WR * WC * 32)
void wmma_gemm_f32(const float* __restrict__ A, long strideA,
                   const float* __restrict__ Bn, int ldb,
                   const float* __restrict__ bias,
                   float* __restrict__ Out, long strideO,
                   int mode)
{
    static_assert(WR * WC == 8, "block must be 8 waves / 256 threads");
    constexpr int BM  = WR * TM * 16;
    constexpr int BN  = WC * TN * 16;
    constexpr int KC  = 32;              // K-chunk staged per buffer
    constexpr int KCP = 36;              // padded row stride: 144B = 9*16 (b128-aligned, bank-spread)
    constexpr int NC  = K / KC;
    constexpr int LA  = BM / 32;         // per-thread b128 DMA issues for the A panel
    constexpr int LB  = BN / 32;         // per-thread b128 DMA issues for the B panel
    constexpr int LPT = LA + LB;

    __shared__ float As[2][BM * KCP];
    __shared__ float Bs[2][BN * KCP];

    const int tid  = threadIdx.x;
    const int lane = tid & 31;
    const int wave = tid >> 5;
    const int l16  = lane & 15;
    const int hi   = lane >> 4;          // 0: lanes 0-15, 1: lanes 16-31
    const int wr   = wave / WC;
    const int wc   = wave % WC;

    const long m0b = (long)blockIdx.x * BM;
    const long n0b = (long)blockIdx.y * BN;

    // Issue the async DMA for K-chunk c into LDS buffer p (whole block cooperates).
    auto issue = [&](int c, int p) {
        const int k0 = c * KC;
#pragma unroll
        for (int r = 0; r < LA; ++r) {
            const int l   = tid + 256 * r;
            const int row = l >> 3;          // 8 b128 segments per 32-float row
            const int seg = l & 7;
            const unsigned lds = (unsigned)(uintptr_t)&As[p][row * KCP + seg * 4];
            async_b128_to_lds(lds, A + (m0b + row) * strideA + k0 + seg * 4);
        }
#pragma unroll
        for (int r = 0; r < LB; ++r) {
            const int l   = tid + 256 * r;
            const int row = l >> 3;
            const int seg = l & 7;
            const unsigned lds = (unsigned)(uintptr_t)&Bs[p][row * KCP + seg * 4];
            async_b128_to_lds(lds, Bn + (long)(n0b + row) * ldb + k0 + seg * 4);
        }
    };

    v8f c[TM][TN];
#pragma unroll
    for (int i = 0; i < TM; ++i)
#pragma unroll
        for (int j = 0; j < TN; ++j)
            c[i][j] = (v8f){0.f, 0.f, 0.f, 0.f, 0.f, 0.f, 0.f, 0.f};

    // Consume one staged K-chunk from LDS buffer p.
    // A frag (16x4 f32): lanes 0-15 K=k,k+1 ; lanes 16-31 K=k+2,k+3  -> ds_load_b64.
    // B frag (4x16 f32): same half-wave split, rows from the N-major panel.
    auto compute = [&](int p) {
#pragma unroll
        for (int kc = 0; kc < KC; kc += 4) {
            v2f a[TM], b[TN];
#pragma unroll
            for (int i = 0; i < TM; ++i)
                a[i] = *(const v2f*)&As[p][(wr * TM * 16 + i * 16 + l16) * KCP + kc + 2 * hi];
#pragma unroll
            for (int j = 0; j < TN; ++j)
                b[j] = *(const v2f*)&Bs[p][(wc * TN * 16 + j * 16 + l16) * KCP + kc + 2 * hi];
#pragma unroll
            for (int i = 0; i < TM; ++i)
#pragma unroll
                for (int j = 0; j < TN; ++j)
                    c[i][j] = __builtin_amdgcn_wmma_f32_16x16x4_f32(
                        false, a[i], false, b[j], (short)0, c[i][j], false, false);
        }
    };

    // Depth-2 software pipeline: DMA for chunk c+1 overlaps WMMA on chunk c.
    issue(0, 0);
    if constexpr (NC > 1) issue(1, 1);
    for (int cc = 0; cc < NC - 1; ++cc) {
        const int p = cc & 1;
        wait_asynccnt<LPT>();            // chunk cc landed (chunk cc+1 may still be in flight)
        __syncthreads();
        compute(p);
        __syncthreads();                 // everyone done reading buffer p
        if (cc + 2 < NC) issue(cc + 2, p);
    }
    wait_asynccnt<0>();
    __syncthreads();
    compute((NC - 1) & 1);

    // Epilogue. C/D layout: VGPR v -> row v (lanes 0-15) / row v+8 (lanes 16-31), col = lane%16.
#pragma unroll
    for (int i = 0; i < TM; ++i) {
#pragma unroll
        for (int j = 0; j < TN; ++j) {
            const long colBase = n0b + wc * TN * 16 + j * 16 + l16;
            const float bv = bias[colBase];
#pragma unroll
            for (int v = 0; v < 8; ++v) {
                const long row = m0b + wr * TM * 16 + i * 16 + v + 8 * hi;
                const long idx = row * strideO + colBase;
                const float acc = c[i][j][v] + bv;
                if (mode == 0) {
                    Out[idx] = acc;
                } else {
                    Out[idx] = tanhf(Out[idx] + acc);
                }
            }
        }
    }
}

// hidden = states[:, T-1, :]
__global__ void rnn_copy_hidden(const float* __restrict__ states, float* __restrict__ hidden)
{
    const int i = blockIdx.x * blockDim.x + threadIdx.x;   // 0 .. B*H-1
    const int b = i / RNN_H;
    const int h = i % RNN_H;
    hidden[i] = states[(long)b * RNN_T * RNN_H + (long)(RNN_T - 1) * RNN_H + h];
}

extern "C" void kernel_launch(void* const* d_in, const int* in_sizes, int n_in,
                              void* d_out, int out_size, void* d_ws, size_t ws_size,
                              hipStream_t stream)
{
    (void)in_sizes; (void)n_in; (void)out_size; (void)d_ws; (void)ws_size;

    const float* input = (const float*)d_in[0];   // [B, T, I]
    const float* h_0   = (const float*)d_in[1];   // [B, H]
    const float* W_ih  = (const float*)d_in[2];   // [H, I]  (N-major for our GEMM)
    const float* W_hh  = (const float*)d_in[3];   // [H, H]
    const float* b_ih  = (const float*)d_in[4];   // [H]
    const float* b_hh  = (const float*)d_in[5];   // [H]

    float* out = (float*)d_out;                   // states [B,T,H] then hidden [B,H]

    const dim3 blk(256);

    // Phase 1: xp = X @ W_ih^T + b_ih. X is [B*T, I] contiguous; out rows at row*H.
    // Wave 32x32 (2x2 tiles), block 64x128 -> grid (512, 8).
    {
        const dim3 g1((RNN_B * RNN_T) / 64, RNN_H / 128);
        wmma_gemm_f32<2, 2, 2, 4, RNN_I><<<g1, blk, 0, stream>>>(
            input, (long)RNN_I,
            W_ih, RNN_I,
            b_ih,
            out, (long)RNN_H,
            /*mode=*/0);
    }

    // Phase 2: sequential scan; stream ordering is the per-step global sync.
    // Wave 16x32 (1x2 tiles), block 64x64 -> grid (1, 16) per step.
    {
        const dim3 g2(1, RNN_H / 64);
        const long TH = (long)RNN_T * RNN_H;
        for (int t = 0; t < RNN_T; ++t) {
            const float* Aptr = (t == 0) ? h_0 : (out + (long)(t - 1) * RNN_H);
            const long   sA   = (t == 0) ? (long)RNN_H : TH;
            wmma_gemm_f32<1, 2, 4, 2, RNN_H><<<g2, blk, 0, stream>>>(
                Aptr, sA,
                W_hh, RNN_H,
                b_hh,
                out + (long)t * RNN_H, TH,
                /*mode=*/1);
        }
    }

    // Phase 3: hidden = states[:, T-1, :].
    rnn_copy_hidden<<<(RNN_B * RNN_H) / 256, 256, 0, stream>>>(
        out, out + (long)RNN_B * RNN_T * RNN_H);
}